// BilateralSlice_62775241998839
// MI455X (gfx1250) — compile-verified
//
#include <hip/hip_runtime.h>
#include <hip/hip_bf16.h>
#include <stdint.h>

// Problem constants (fixed by the reference setup_inputs)
#define BATCH 4
#define NCH   12
#define DG    8
#define HG    16
#define WG    16
#define IMH   1024
#define IMW   1024

#define THREADS        256
#define ROWS_PER_BLOCK 16
#define GRID_PER_B     (NCH * DG * HG * WG)   // 24576 floats = 96 KB
#define CH_STRIDE      (DG * HG * WG)         // 2048 floats per channel

typedef int   v4i __attribute__((ext_vector_type(4)));
typedef float v4f __attribute__((ext_vector_type(4)));

__global__ __launch_bounds__(THREADS)
void bilateral_slice_kernel(const float* __restrict__ grid,
                            const float* __restrict__ guide,
                            float* __restrict__ out)
{
    __shared__ __align__(16) float sg[GRID_PER_B];   // one batch's grid, 96 KB

    const int blocksPerBatch = IMH / ROWS_PER_BLOCK; // 64
    const int b  = blockIdx.x / blocksPerBatch;
    const int yb = (blockIdx.x % blocksPerBatch) * ROWS_PER_BLOCK;
    const int t  = threadIdx.x;

    // ---- stage this batch's bilateral grid into LDS (async DMA path) ----
    const float* gsrc = grid + (size_t)b * GRID_PER_B;
#if __has_builtin(__builtin_amdgcn_global_load_async_to_lds_b128)
    {
        typedef __attribute__((address_space(1))) v4i gv4i;
        typedef __attribute__((address_space(3))) v4i lv4i;
        #pragma unroll
        for (int i = 0; i < GRID_PER_B / (4 * THREADS); ++i) {   // 24 x b128 per lane
            const int j = (t + i * THREADS) * 4;                  // float index, 16B aligned
            __builtin_amdgcn_global_load_async_to_lds_b128(
                (gv4i*)(uintptr_t)(gsrc + j),
                (lv4i*)(uint32_t)(uintptr_t)(sg + j),
                0, 0);
        }
    }
  #if __has_builtin(__builtin_amdgcn_s_wait_asynccnt)
    __builtin_amdgcn_s_wait_asynccnt(0);
  #else
    asm volatile("s_wait_asynccnt 0" ::: "memory");
  #endif
#else
    #pragma unroll
    for (int i = 0; i < GRID_PER_B / (4 * THREADS); ++i) {
        const int j = (t + i * THREADS) * 4;
        *(float4*)(sg + j) = *(const float4*)(gsrc + j);
    }
#endif
    __syncthreads();

    // ---- per-pixel trilinear slice -------------------------------------
    const float scale = (float)(WG - 1) / (float)(IMW - 1);  // == 15/1023 (HG/IMH identical)
    const int x4 = t * 4;

    float res[NCH][4];

    for (int r = 0; r < ROWS_PER_BLOCK; ++r) {
        const int y = yb + r;

        const float iy = (float)y * scale;
        const float py = floorf(iy);
        const float fy = iy - py;
        const int   yA = (int)py;
        const int   yB = min(yA + 1, HG - 1);
        const float wy0 = 1.0f - fy, wy1 = fy;

        // read-once guide: non-temporal load (don't pollute L2)
        const v4f g4 = __builtin_nontemporal_load(
            (const v4f*)(guide + ((size_t)b * IMH + y) * IMW + x4));
        const float gz[4] = { g4.x, g4.y, g4.z, g4.w };

        #pragma unroll
        for (int j = 0; j < 4; ++j) {
            const float ix = (float)(x4 + j) * scale;
            const float px = floorf(ix);
            const float fx = ix - px;
            const int   xA = (int)px;
            const int   xB = min(xA + 1, WG - 1);
            const float wx0 = 1.0f - fx, wx1 = fx;

            const float z  = fminf(fmaxf(gz[j], 0.0f), 1.0f) * (float)(DG - 1);
            const float pz = floorf(z);
            const float fz = z - pz;
            const int   zA = (int)pz;
            const int   zB = min(zA + 1, DG - 1);
            const float wz0 = 1.0f - fz, wz1 = fz;

            const int rowA = yA * WG, rowB = yB * WG;
            const int plA  = zA * (HG * WG), plB = zB * (HG * WG);
            const int i000 = plA + rowA + xA;
            const int i001 = plA + rowA + xB;
            const int i010 = plA + rowB + xA;
            const int i011 = plA + rowB + xB;
            const int i100 = plB + rowA + xA;
            const int i101 = plB + rowA + xB;
            const int i110 = plB + rowB + xA;
            const int i111 = plB + rowB + xB;

            #pragma unroll
            for (int c = 0; c < NCH; ++c) {
                const float* s = sg + c * CH_STRIDE;
                const float v00 = s[i000] * wx0 + s[i001] * wx1;
                const float v01 = s[i010] * wx0 + s[i011] * wx1;
                const float v10 = s[i100] * wx0 + s[i101] * wx1;
                const float v11 = s[i110] * wx0 + s[i111] * wx1;
                const float v0  = v00 * wy0 + v01 * wy1;
                const float v1  = v10 * wy0 + v11 * wy1;
                res[c][j] = v0 * wz0 + v1 * wz1;
            }
        }

        // write-once output: non-temporal b128 stores
        #pragma unroll
        for (int c = 0; c < NCH; ++c) {
            v4f o;
            o.x = res[c][0]; o.y = res[c][1]; o.z = res[c][2]; o.w = res[c][3];
            __builtin_nontemporal_store(
                o, (v4f*)(out + (((size_t)b * NCH + c) * IMH + y) * IMW + x4));
        }
    }
}

extern "C" void kernel_launch(void* const* d_in, const int* in_sizes, int n_in,
                              void* d_out, int out_size, void* d_ws, size_t ws_size,
                              hipStream_t stream)
{
    (void)in_sizes; (void)n_in; (void)out_size; (void)d_ws; (void)ws_size;
    const float* grid  = (const float*)d_in[0];
    const float* guide = (const float*)d_in[1];
    float* out = (float*)d_out;

    dim3 block(THREADS);
    dim3 grd(BATCH * (IMH / ROWS_PER_BLOCK));   // 256 blocks
    hipLaunchKernelGGL(bilateral_slice_kernel, grd, block, 0, stream, grid, guide, out);
}